// Ergodicity_Loss_6150393168131
// MI455X (gfx1250) — compile-verified
//
#include <hip/hip_runtime.h>

typedef float v2f __attribute__((ext_vector_type(2)));
typedef float v8f __attribute__((ext_vector_type(8)));

#define T_DIM 512
#define B_DIM 32
#define N_DIM 64
#define K_DIM 32
#define CHUNKS 16
#define WAVES 8
#define SAMPLES_PER_WAVE ((T_DIM * N_DIM) / (CHUNKS * WAVES)) // 256
#define GROUPS (SAMPLES_PER_WAVE / 4)                          // 64

// ---------------------------------------------------------------------------
// Main kernel: per-batch S = C0^T * C1 via V_WMMA_F32_16X16X4_F32.
// Each wave owns a 32x32 f32 accumulator (4 tiles of 16x16 = 32 VGPRs) and
// streams 256 samples, 4 per WMMA K-step. Cosines feed the TRANS pipe while
// WMMA runs on the matrix pipe.
// A-matrix layout (f32 16x4): lanes 0-15 -> K=0(v0),K=1(v1); lanes 16-31 ->
// K=2(v0),K=3(v1); M = lane%16. B mirrored with N = lane%16.
// ---------------------------------------------------------------------------
__global__ __launch_bounds__(256) void ergo_main(const float* __restrict__ x,
                                                 const float* __restrict__ L,
                                                 float* __restrict__ gPart)
{
    const int b     = blockIdx.x;
    const int chunk = blockIdx.y;
    const int wave  = threadIdx.x >> 5;
    const int lane  = threadIdx.x & 31;
    const int half  = lane >> 4;
    const int m     = lane & 15;

    const float f0  = 3.14159265358979323846f / L[0];
    const float f1  = 3.14159265358979323846f / L[1];
    const float mLo = (float)m;
    const float mHi = (float)(m + 16);

    v8f acc00 = {}; v8f acc01 = {}; v8f acc10 = {}; v8f acc11 = {};

    const int sampleBase = (chunk * WAVES + wave) * SAMPLES_PER_WAVE;

    for (int g = 0; g < GROUPS; ++g) {
        const int s0 = sampleBase + g * 4 + half * 2;
        v2f aLo, aHi, bLo, bHi;
#pragma unroll
        for (int j = 0; j < 2; ++j) {
            const int s    = s0 + j;
            const int t    = s >> 6;   // s / N_DIM
            const int n    = s & 63;   // s % N_DIM
            const int addr = ((t * B_DIM + b) * N_DIM + n) * 2;
            const float x0 = x[addr];
            const float x1 = x[addr + 1];
            if (j == 0) {
                // speculative prefetch of next group's sample for this lane
                const int s2 = s + 4;
                const int a2 = (((s2 >> 6) * B_DIM + b) * N_DIM + (s2 & 63)) * 2;
                __builtin_prefetch(x + a2, 0, 0);
            }
            aLo[j] = __cosf(x0 * f0 * mLo);
            aHi[j] = __cosf(x0 * f0 * mHi);
            bLo[j] = __cosf(x1 * f1 * mLo);
            bHi[j] = __cosf(x1 * f1 * mHi);
        }
        // D = A x B + C, f32 throughout (faithful to the f32 reference)
        acc00 = __builtin_amdgcn_wmma_f32_16x16x4_f32(false, aLo, false, bLo,
                                                      (short)0, acc00, false, false);
        acc01 = __builtin_amdgcn_wmma_f32_16x16x4_f32(false, aLo, false, bHi,
                                                      (short)0, acc01, false, false);
        acc10 = __builtin_amdgcn_wmma_f32_16x16x4_f32(false, aHi, false, bLo,
                                                      (short)0, acc10, false, false);
        acc11 = __builtin_amdgcn_wmma_f32_16x16x4_f32(false, aHi, false, bHi,
                                                      (short)0, acc11, false, false);
    }

    // C/D layout: VGPR r -> M=r (lanes 0-15) / M=r+8 (lanes 16-31); N = lane%16.
    __shared__ float sS[WAVES * K_DIM * K_DIM];
    float* myS = sS + wave * (K_DIM * K_DIM);
#pragma unroll
    for (int r = 0; r < 8; ++r) {
        const int p = r + 8 * half;
        myS[p * K_DIM + m]               = acc00[r];
        myS[p * K_DIM + m + 16]          = acc01[r];
        myS[(p + 16) * K_DIM + m]        = acc10[r];
        myS[(p + 16) * K_DIM + m + 16]   = acc11[r];
    }
    __syncthreads();

    // Deterministic per-workgroup partial (no float atomics).
    const int wg = b * CHUNKS + chunk;
#pragma unroll
    for (int k = 0; k < 4; ++k) {
        const int e = threadIdx.x * 4 + k;
        float sum = 0.f;
#pragma unroll
        for (int w = 0; w < WAVES; ++w) sum += sS[w * (K_DIM * K_DIM) + e];
        gPart[wg * (K_DIM * K_DIM) + e] = sum;
    }
}

// ---------------------------------------------------------------------------
// Memory-bound sum of u^2 -> 256 deterministic block partials.
// ---------------------------------------------------------------------------
__global__ __launch_bounds__(256) void u2_partial(const float* __restrict__ u,
                                                  float* __restrict__ uPart, int n)
{
    float acc = 0.f;
    for (int i = blockIdx.x * blockDim.x + threadIdx.x; i < n;
         i += gridDim.x * blockDim.x) {
        const float v = u[i];
        acc += v * v;
    }
    __shared__ float red[256];
    red[threadIdx.x] = acc;
    __syncthreads();
    for (int s = 128; s > 0; s >>= 1) {
        if (threadIdx.x < (unsigned)s) red[threadIdx.x] += red[threadIdx.x + s];
        __syncthreads();
    }
    if (threadIdx.x == 0) uPart[blockIdx.x] = red[0];
}

// ---------------------------------------------------------------------------
// Finalize: coeffs = S/(nf*N*T); loss = sum((coeffs-cd)^2 * nw) + reg.
// ---------------------------------------------------------------------------
__global__ __launch_bounds__(1024) void finalize(const float* __restrict__ gPart,
                                                 const float* __restrict__ uPart,
                                                 const float* __restrict__ cd,
                                                 const float* __restrict__ nf,
                                                 const float* __restrict__ nw,
                                                 float* __restrict__ out)
{
    const int e = threadIdx.x;          // one (p,q) entry per thread, 1024 total
    const float nfv = nf[e];
    const float cdv = cd[e];
    const float nwv = nw[e];
    const float invNT = 1.0f / (float)(N_DIM * T_DIM);

    float acc = 0.f;
    for (int b = 0; b < B_DIM; ++b) {
        float S = 0.f;
        for (int c = 0; c < CHUNKS; ++c)
            S += gPart[(b * CHUNKS + c) * (K_DIM * K_DIM) + e];
        const float coeff = S * invNT / nfv - cdv;
        acc += coeff * coeff * nwv;
    }

    __shared__ float red[1024];
    red[e] = acc;
    __syncthreads();
    for (int s = 512; s > 0; s >>= 1) {
        if (e < s) red[e] += red[e + s];
        __syncthreads();
    }
    if (e == 0) {
        float us = 0.f;
        for (int i = 0; i < 256; ++i) us += uPart[i];
        out[0] = red[0] + 1e-3f * us / (2.0f * N_DIM * T_DIM * B_DIM);
    }
}

extern "C" void kernel_launch(void* const* d_in, const int* in_sizes, int n_in,
                              void* d_out, int out_size, void* d_ws, size_t ws_size,
                              hipStream_t stream)
{
    const float* x  = (const float*)d_in[0];  // [T,B,N,2]
    const float* u  = (const float*)d_in[1];  // [T,B,N,2]
    const float* L  = (const float*)d_in[2];  // [2]
    const float* cd = (const float*)d_in[3];  // [32,32]
    const float* nf = (const float*)d_in[4];  // [32,32]
    const float* nw = (const float*)d_in[5];  // [32,32]
    float* out = (float*)d_out;

    float* gPart = (float*)d_ws;                                // 512*1024 f32
    float* uPart = gPart + (size_t)B_DIM * CHUNKS * K_DIM * K_DIM; // 256 f32

    ergo_main<<<dim3(B_DIM, CHUNKS), 256, 0, stream>>>(x, L, gPart);
    u2_partial<<<256, 256, 0, stream>>>(u, uPart, T_DIM * B_DIM * N_DIM * 2);
    finalize<<<1, 1024, 0, stream>>>(gPart, uPart, cd, nf, nw, out);
}